// DeepGAT_69690139344897
// MI455X (gfx1250) — compile-verified
//
#include <hip/hip_runtime.h>
#include <math.h>

typedef __attribute__((ext_vector_type(2))) float v2f;
typedef __attribute__((ext_vector_type(4))) float v4f;
typedef __attribute__((ext_vector_type(8))) float v8f;
typedef __attribute__((ext_vector_type(4))) unsigned int v4u;
typedef __attribute__((ext_vector_type(8))) int v8i;
typedef __attribute__((ext_vector_type(4))) int v4i;

#define NEG_SLOPE 0.2f

#if defined(__AMDGCN__) && __has_builtin(__builtin_amdgcn_tensor_load_to_lds) && \
    __has_builtin(__builtin_amdgcn_s_wait_tensorcnt)
#define USE_TDM 1
#else
#define USE_TDM 0
#endif

__device__ __forceinline__ void atomicMaxF(float* addr, float val) {
    // classic signed/unsigned trick; memory initialized to -INF (0xFF800000)
    if (val >= 0.0f) atomicMax((int*)addr, __float_as_int(val));
    else             atomicMin((unsigned int*)addr, __float_as_uint(val));
}

__global__ void zero_f32(float* __restrict__ p, long long n) {
    long long i = (long long)blockIdx.x * blockDim.x + threadIdx.x;
    if (i < n) p[i] = 0.0f;
}

__global__ void init_softmax_state(float* __restrict__ m, float* __restrict__ den, int n) {
    int i = blockIdx.x * blockDim.x + threadIdx.x;
    if (i < n) { m[i] = -INFINITY; den[i] = 0.0f; }
}

#if USE_TDM
// One-shot TDM DMA of a 2D f32 tile (tile0 x tile1 dwords) into LDS, with
// pad_amount=4 dwords appended per pad-interval so LDS rows are (tile0+4) floats.
// Rows beyond tensor_dim1 are zero-filled by the TDM OOB rule.
__device__ __forceinline__ void tdm_load_2d(float* lds_dst, const float* gsrc,
                                            unsigned dim0, unsigned dim1,
                                            unsigned tile0, unsigned tile1,
                                            unsigned stride0, unsigned pad_code) {
    unsigned lds = (unsigned)(size_t)lds_dst;           // low 32 bits = LDS offset
    unsigned long long ga = (unsigned long long)(size_t)gsrc;
    v4u g0;
    g0.x = 1u;                                          // count=1, is_restore=0, gather=0
    g0.y = lds;                                         // lds_addr [63:32]
    g0.z = (unsigned)(ga & 0xFFFFFFFFu);                // global_addr [95:64]
    g0.w = (unsigned)(ga >> 32) | (2u << 30);           // global_addr hi + type=2
    // group1: data_size=4B(2), pad_enable=1, pad_interval=pad_code, pad_amount code 3 (=4 dwords)
    unsigned d0 = (2u << 16) | (1u << 20) | (pad_code << 22) | (3u << 25);
    unsigned d1 = (dim0 & 0xFFFFu) << 16;               // tensor_dim0 [79:48]
    unsigned d2 = (dim0 >> 16) | ((dim1 & 0xFFFFu) << 16); // dim0 hi | tensor_dim1 [111:80]
    unsigned d3 = (dim1 >> 16) | (tile0 << 16);         // dim1 hi | tile_dim0 [127:112]
    unsigned d4 = tile1;                                // tile_dim1 [143:128], tile_dim2=0
    unsigned d5 = stride0;                              // tensor_dim0_stride [207:160] low
    v8i g1 = { (int)d0, (int)d1, (int)d2, (int)d3, (int)d4, (int)d5, 0, 0 };
    v4i z4 = { 0, 0, 0, 0 };
    v8i z8 = { 0, 0, 0, 0, 0, 0, 0, 0 };
    // this toolchain declares the 6-arg form: (g0, g1, g2, g3, g4, cpol)
    __builtin_amdgcn_tensor_load_to_lds(g0, g1, z4, z4, z8, 0);
}
#endif

// H[M,NC] = X[M,128] @ W[128,NC] (f32, row-major).
// Block = (NC/16) waves; block computes a 32-row x NC strip.
// A strip (32x128) and W (128xNC) are staged in LDS (TDM when available),
// rows padded by 4 dwords for conflict-free ds reads. Each wave owns one
// 16-column tile and keeps TWO 16x16 f32 accumulators (rows r..r+15 and
// r+16..r+31) sharing each B fragment -> 2 independent WMMA chains.
template <int NC>
__global__ void gemm_wmma_f32(const float* __restrict__ X, const float* __restrict__ W,
                              float* __restrict__ H, int M) {
    constexpr int K    = 128;
    constexpr int WROW = NC + 4;          // padded W row (floats)
    constexpr int AROW = K + 4;           // padded A row (floats)
    constexpr int NTHR = (NC / 16) * 32;  // threads per block
    constexpr unsigned WPAD_CODE = (NC == 128) ? 6u : 5u;  // 128 / 64 dwords per row
    __shared__ float Ws[K * WROW];
    __shared__ float As[32 * AROW];

    const int tid  = threadIdx.x;
    const int lane = tid & 31;
    const int wv   = tid >> 5;            // tile_n within the block
    const int row0 = blockIdx.x * 32;

#if USE_TDM
    if (tid < 32) {
        tdm_load_2d(Ws, W, NC, K, NC, K, NC, WPAD_CODE);
        unsigned rows = (unsigned)((M - row0) < 32 ? (M - row0) : 32);
        tdm_load_2d(As, X + (size_t)row0 * K, K, rows, K, 32, K, 6u);
        __builtin_amdgcn_s_wait_tensorcnt((short)0);
    }
    __syncthreads();
#else
    for (int idx = tid; idx < K * (NC / 4); idx += NTHR) {
        int kr = idx / (NC / 4), cc = idx % (NC / 4);
        *(v4f*)&Ws[kr * WROW + cc * 4] = *(const v4f*)&W[(size_t)kr * NC + cc * 4];
    }
    for (int idx = tid; idx < 32 * (K / 4); idx += NTHR) {
        int r = idx / (K / 4), cc = idx % (K / 4);
        int gr = row0 + r;
        v4f val = {};
        if (gr < M) val = *(const v4f*)&X[(size_t)gr * K + cc * 4];
        *(v4f*)&As[r * AROW + cc * 4] = val;
    }
    __syncthreads();
#endif

    const int hl  = lane >> 4;            // 0: K pair (k,k+1); 1: (k+2,k+3)
    const int l16 = lane & 15;
    const float* a0p  = &As[l16 * AROW + 2 * hl];
    const float* a1p  = &As[(16 + l16) * AROW + 2 * hl];
    const float* wcol = &Ws[(2 * hl) * WROW + wv * 16 + l16];
    v8f acc0 = {}, acc1 = {};
#pragma unroll 4
    for (int k = 0; k < K; k += 4) {
        v2f b;
        b.x = wcol[k * WROW];
        b.y = wcol[(k + 1) * WROW];
        v2f a0 = *(const v2f*)&a0p[k];
        v2f a1 = *(const v2f*)&a1p[k];
        acc0 = __builtin_amdgcn_wmma_f32_16x16x4_f32(false, a0, false, b,
                                                     (short)0, acc0, false, false);
        acc1 = __builtin_amdgcn_wmma_f32_16x16x4_f32(false, a1, false, b,
                                                     (short)0, acc1, false, false);
    }
    // C/D layout: lanes 0-15 hold rows 0..7, lanes 16-31 hold rows 8..15
    const int col = wv * 16 + l16;
#pragma unroll
    for (int r = 0; r < 8; ++r) {
        int r0 = row0 + hl * 8 + r;
        int r1 = r0 + 16;
        if (r0 < M) H[(size_t)r0 * NC + col] = acc0[r];
        if (r1 < M) H[(size_t)r1 * NC + col] = acc1[r];
    }
}

// al[n,h] = <h[n,h,:], a_src[h,:]> ; ar likewise. One thread per (node, head).
__global__ void attn_coef(const float* __restrict__ H,
                          const float* __restrict__ a_src, const float* __restrict__ a_dst,
                          float* __restrict__ al, float* __restrict__ ar,
                          int n, int n_head, int head_dim) {
    int i = blockIdx.x * blockDim.x + threadIdx.x;
    if (i >= n * n_head) return;
    int node = i / n_head;
    int h    = i % n_head;
    const float* hp = H + (size_t)node * n_head * head_dim + (size_t)h * head_dim;
    const float* as = a_src + (size_t)h * head_dim;
    const float* ad = a_dst + (size_t)h * head_dim;
    float sl = 0.0f, sr = 0.0f;
    for (int c = 0; c < head_dim; ++c) {
        float v = hp[c];
        sl += v * as[c];
        sr += v * ad[c];
    }
    al[i] = sl;
    ar[i] = sr;
}

// segment_max over dst of leaky_relu(al[src]+ar[dst]); one thread per edge.
__global__ void edge_max(const int* __restrict__ src, const int* __restrict__ dst,
                         int Ereal, int n,
                         const float* __restrict__ al, const float* __restrict__ ar,
                         float* __restrict__ m, int n_head) {
    int e = blockIdx.x * blockDim.x + threadIdx.x;
    int total = Ereal + n;
    if (e >= total) return;
    int s = (e < Ereal) ? src[e] : (e - Ereal);   // implicit self-loops
    int d = (e < Ereal) ? dst[e] : (e - Ereal);
    for (int h = 0; h < n_head; ++h) {
        float v = al[(size_t)s * n_head + h] + ar[(size_t)d * n_head + h];
        v = (v > 0.0f) ? v : NEG_SLOPE * v;
        atomicMaxF(&m[(size_t)d * n_head + h], v);
    }
}

// den[dst,h] += exp(e - m[dst,h])
__global__ void edge_expsum(const int* __restrict__ src, const int* __restrict__ dst,
                            int Ereal, int n,
                            const float* __restrict__ al, const float* __restrict__ ar,
                            const float* __restrict__ m, float* __restrict__ den, int n_head) {
    int e = blockIdx.x * blockDim.x + threadIdx.x;
    int total = Ereal + n;
    if (e >= total) return;
    int s = (e < Ereal) ? src[e] : (e - Ereal);
    int d = (e < Ereal) ? dst[e] : (e - Ereal);
    for (int h = 0; h < n_head; ++h) {
        float v = al[(size_t)s * n_head + h] + ar[(size_t)d * n_head + h];
        v = (v > 0.0f) ? v : NEG_SLOPE * v;
        float ee = expf(v - m[(size_t)d * n_head + h]);
        atomicAdd(&den[(size_t)d * n_head + h], ee);
    }
}

// out[dst,h,c] += h[src,h,c] * alpha ; one wave (32 lanes) per edge.
template <int CH, int NHEAD>
__global__ void edge_scatter(const int* __restrict__ src, const int* __restrict__ dst,
                             int Ereal, int n,
                             const float* __restrict__ H,
                             const float* __restrict__ al, const float* __restrict__ ar,
                             const float* __restrict__ m, const float* __restrict__ den,
                             float* __restrict__ out) {
    constexpr int CPL = CH / 32;      // channels per lane (4 or 2)
    constexpr int HD  = CH / NHEAD;   // head dim
    int wid  = (blockIdx.x * blockDim.x + threadIdx.x) >> 5;
    int lane = threadIdx.x & 31;
    int total = Ereal + n;
    if (wid >= total) return;
    int s = (wid < Ereal) ? src[wid] : (wid - Ereal);
    int d = (wid < Ereal) ? dst[wid] : (wid - Ereal);
    int c0 = lane * CPL;
    int h  = c0 / HD;
    float e = al[(size_t)s * NHEAD + h] + ar[(size_t)d * NHEAD + h];
    e = (e > 0.0f) ? e : NEG_SLOPE * e;
    float alpha = expf(e - m[(size_t)d * NHEAD + h]) /
                  (den[(size_t)d * NHEAD + h] + 1e-16f);
    const float* hp = H + (size_t)s * CH + c0;
    float* op = out + (size_t)d * CH + c0;
    float hv[CPL];
    if (CPL == 4) *(v4f*)hv = *(const v4f*)hp;
    else          *(v2f*)hv = *(const v2f*)hp;
#pragma unroll
    for (int c = 0; c < CPL; ++c) atomicAdd(op + c, hv[c] * alpha);
}

// y = LN(x + gat_bias) * g + be, optional ELU. One wave per node, in place.
template <int CH, bool DO_ELU>
__global__ void bias_ln_act(float* __restrict__ X, const float* __restrict__ bias,
                            const float* __restrict__ g, const float* __restrict__ be, int n) {
    constexpr int CPL = CH / 32;
    int wid  = (blockIdx.x * blockDim.x + threadIdx.x) >> 5;
    int lane = threadIdx.x & 31;
    if (wid >= n) return;
    float* xp = X + (size_t)wid * CH + lane * CPL;
    float v[CPL];
    float s = 0.0f;
#pragma unroll
    for (int c = 0; c < CPL; ++c) { v[c] = xp[c] + bias[lane * CPL + c]; s += v[c]; }
#pragma unroll
    for (int o = 16; o > 0; o >>= 1) s += __shfl_xor(s, o, 32);
    float mu = s * (1.0f / CH);
    float var = 0.0f;
#pragma unroll
    for (int c = 0; c < CPL; ++c) { float dv = v[c] - mu; var += dv * dv; }
#pragma unroll
    for (int o = 16; o > 0; o >>= 1) var += __shfl_xor(var, o, 32);
    var *= (1.0f / CH);
    float rs = rsqrtf(var + 1e-5f);
#pragma unroll
    for (int c = 0; c < CPL; ++c) {
        float y = (v[c] - mu) * rs * g[lane * CPL + c] + be[lane * CPL + c];
        if (DO_ELU) y = (y > 0.0f) ? y : expm1f(y);
        xp[c] = y;
    }
}

static inline int cdiv(long long a, long long b) { return (int)((a + b - 1) / b); }

extern "C" void kernel_launch(void* const* d_in, const int* in_sizes, int n_in,
                              void* d_out, int out_size, void* d_ws, size_t ws_size,
                              hipStream_t stream) {
    const float* x   = (const float*)d_in[0];
    const int*   ei  = (const int*)d_in[1];
    const float* W1  = (const float*)d_in[2];
    const float* as1 = (const float*)d_in[3];
    const float* ad1 = (const float*)d_in[4];
    const float* b1  = (const float*)d_in[5];
    const float* g1  = (const float*)d_in[6];
    const float* be1 = (const float*)d_in[7];
    const float* W2  = (const float*)d_in[8];
    const float* as2 = (const float*)d_in[9];
    const float* ad2 = (const float*)d_in[10];
    const float* b2  = (const float*)d_in[11];
    const float* g2  = (const float*)d_in[12];
    const float* be2 = (const float*)d_in[13];
    const float* W3  = (const float*)d_in[14];
    const float* as3 = (const float*)d_in[15];
    const float* ad3 = (const float*)d_in[16];
    const float* b3  = (const float*)d_in[17];
    const float* g3  = (const float*)d_in[18];
    const float* be3 = (const float*)d_in[19];

    const int N = in_sizes[0] / 128;
    const int E = in_sizes[1] / 2;
    const int* srcv = ei;
    const int* dstv = ei + E;
    const int total = E + N;

    // workspace layout (floats): A=h buffer, B=x/out ping buffer, + per-head arrays
    float* A   = (float*)d_ws;                 // N*128
    float* B   = A + (size_t)N * 128;          // N*128
    float* al  = B + (size_t)N * 128;          // N*8
    float* ar  = al + (size_t)N * 8;           // N*8
    float* m   = ar + (size_t)N * 8;           // N*8
    float* den = m  + (size_t)N * 8;           // N*8

    const dim3 blk(256);
    const int gemm_blocks = cdiv(N, 32);

    // ---------------- Layer 1: x[N,128] -> B[N,128] ----------------
    {
        gemm_wmma_f32<128><<<gemm_blocks, dim3(256), 0, stream>>>(x, W1, A, N);
        attn_coef<<<cdiv((long long)N * 8, 256), blk, 0, stream>>>(A, as1, ad1, al, ar, N, 8, 16);
        init_softmax_state<<<cdiv((long long)N * 8, 256), blk, 0, stream>>>(m, den, N * 8);
        edge_max<<<cdiv(total, 256), blk, 0, stream>>>(srcv, dstv, E, N, al, ar, m, 8);
        edge_expsum<<<cdiv(total, 256), blk, 0, stream>>>(srcv, dstv, E, N, al, ar, m, den, 8);
        zero_f32<<<cdiv((long long)N * 128, 256), blk, 0, stream>>>(B, (long long)N * 128);
        edge_scatter<128, 8><<<cdiv((long long)total * 32, 256), blk, 0, stream>>>(
            srcv, dstv, E, N, A, al, ar, m, den, B);
        bias_ln_act<128, true><<<cdiv((long long)N * 32, 256), blk, 0, stream>>>(B, b1, g1, be1, N);
    }

    // ---------------- Layer 2: B[N,128] -> B[N,128] ----------------
    {
        gemm_wmma_f32<128><<<gemm_blocks, dim3(256), 0, stream>>>(B, W2, A, N);
        attn_coef<<<cdiv((long long)N * 8, 256), blk, 0, stream>>>(A, as2, ad2, al, ar, N, 8, 16);
        init_softmax_state<<<cdiv((long long)N * 8, 256), blk, 0, stream>>>(m, den, N * 8);
        edge_max<<<cdiv(total, 256), blk, 0, stream>>>(srcv, dstv, E, N, al, ar, m, 8);
        edge_expsum<<<cdiv(total, 256), blk, 0, stream>>>(srcv, dstv, E, N, al, ar, m, den, 8);
        // B (old x) is dead after the GEMM above -> reuse as the output accumulator
        zero_f32<<<cdiv((long long)N * 128, 256), blk, 0, stream>>>(B, (long long)N * 128);
        edge_scatter<128, 8><<<cdiv((long long)total * 32, 256), blk, 0, stream>>>(
            srcv, dstv, E, N, A, al, ar, m, den, B);
        bias_ln_act<128, true><<<cdiv((long long)N * 32, 256), blk, 0, stream>>>(B, b2, g2, be2, N);
    }

    // ---------------- Layer 3: B[N,128] -> d_out[N,64], 1 head, no ELU ----------------
    {
        float* out = (float*)d_out;
        gemm_wmma_f32<64><<<gemm_blocks, dim3(128), 0, stream>>>(B, W3, A, N);
        attn_coef<<<cdiv((long long)N, 256), blk, 0, stream>>>(A, as3, ad3, al, ar, N, 1, 64);
        init_softmax_state<<<cdiv((long long)N, 256), blk, 0, stream>>>(m, den, N);
        edge_max<<<cdiv(total, 256), blk, 0, stream>>>(srcv, dstv, E, N, al, ar, m, 1);
        edge_expsum<<<cdiv(total, 256), blk, 0, stream>>>(srcv, dstv, E, N, al, ar, m, den, 1);
        zero_f32<<<cdiv((long long)N * 64, 256), blk, 0, stream>>>(out, (long long)N * 64);
        edge_scatter<64, 1><<<cdiv((long long)total * 32, 256), blk, 0, stream>>>(
            srcv, dstv, E, N, A, al, ar, m, den, out);
        bias_ln_act<64, false><<<cdiv((long long)N * 32, 256), blk, 0, stream>>>(out, b3, g3, be3, N);
    }
}